// HybridRetention_6605659701981
// MI455X (gfx1250) — compile-verified
//
#include <hip/hip_runtime.h>
#include <math.h>

// Problem constants (match reference)
#define BB 4
#define TT 2048
#define DD 512
#define HH 8
#define DH 64
#define CL 64      // chunk length (retention + scan)
#define NCHUNK (TT / CL)
#define PITCH 68   // LDS row pitch (floats): 68 mod 64 = 4 -> conflict-free column reads
                   // 68*4 = 272 bytes = 17*16 -> every 4-float group stays 16B-aligned

typedef __attribute__((ext_vector_type(2))) float v2f;
typedef __attribute__((ext_vector_type(8))) float v8f;

// D = A(16x4, f32) * B(4x16, f32) + C(16x16, f32)
__device__ __forceinline__ v8f wmma4(v2f a, v2f b, v8f c) {
  return __builtin_amdgcn_wmma_f32_16x16x4_f32(
      /*neg_a=*/false, a, /*neg_b=*/false, b,
      /*c_mod=*/(short)0, c, /*reuse_a=*/false, /*reuse_b=*/false);
}

// Low 32 bits of a flat pointer into the LDS aperture == wave-relative LDS addr
// (see ISA 10.2).
__device__ __forceinline__ uint32_t lds_addr32(const void* p) {
  return (uint32_t)(unsigned long long)p;
}

// CDNA5 async global->LDS copy: each lane moves 16 bytes, tracked by ASYNCcnt.
__device__ __forceinline__ void async_b128(uint32_t lds_byte, const float* src) {
  asm volatile("global_load_async_to_lds_b128 %0, %1, off"
               :: "v"(lds_byte), "v"(src) : "memory");
}
__device__ __forceinline__ void wait_async0() {
  asm volatile("s_wait_asynccnt 0" ::: "memory");
}

// A fragment (16x4 tile at rows [mbase,mbase+16), cols [kk,kk+4)):
// lane L holds A[mbase + L%16][kk + 2*(L/16) + {0,1}]
__device__ __forceinline__ v2f ldA(const float* t, int mbase, int kk, int lane) {
  int m = mbase + (lane & 15);
  int k = kk + 2 * (lane >> 4);
  v2f a; a.x = t[m * PITCH + k]; a.y = t[m * PITCH + k + 1]; return a;
}
// B fragment (4x16 tile): lane L holds B[kk + 2*(L/16) + {0,1}][nbase + L%16]
__device__ __forceinline__ v2f ldB(const float* t, int nbase, int kk, int lane) {
  int n = nbase + (lane & 15);
  int k = kk + 2 * (lane >> 4);
  v2f b; b.x = t[k * PITCH + n]; b.y = t[(k + 1) * PITCH + n]; return b;
}
// B fragment reading a transposed tile: B[k][n] = t[n][k]  (for Q @ K^T)
__device__ __forceinline__ v2f ldBT(const float* t, int nbase, int kk, int lane) {
  int n = nbase + (lane & 15);
  int k = kk + 2 * (lane >> 4);
  v2f b; b.x = t[n * PITCH + k]; b.y = t[n * PITCH + k + 1]; return b;
}

// ---------------------------------------------------------------------------
// GEMM: out[M=8192, N=512] = act(X[8192,512] @ W[512,512] + bias) (+ add)
// 64x64 tile per 256-thread block; 8 waves each own two 16x16 accumulators.
// K-panels staged into LDS with async global->LDS (no VGPR round trip).
// ---------------------------------------------------------------------------
__global__ __launch_bounds__(256) void gemm512_kernel(
    const float* X, const float* W, const float* bias,
    float* out, const float* add, int sig)
{
  __shared__ float Xs[64 * PITCH];
  __shared__ float Ws[64 * PITCH];
  const int tid = threadIdx.x, lane = tid & 31, wave = tid >> 5;
  const int m0 = blockIdx.x * 64, n0 = blockIdx.y * 64;
  const int mb = 16 * (wave & 3), nb0 = 32 * (wave >> 2), nb1 = nb0 + 16;
  const uint32_t xs_b = lds_addr32(Xs), ws_b = lds_addr32(Ws);

  v8f c0 = {}, c1 = {};
  for (int kb = 0; kb < DD; kb += 64) {
    // 64x64 f32 tile = 1024 16B groups per array; 4 per thread.
#pragma unroll
    for (int it = 0; it < 4; ++it) {
      int e = tid + 256 * it;
      int r = e >> 4, c4 = (e & 15) << 2;
      uint32_t l = (uint32_t)((r * PITCH + c4) * 4);
      async_b128(xs_b + l, X + (size_t)(m0 + r) * DD + kb + c4);
      async_b128(ws_b + l, W + (size_t)(kb + r) * DD + n0 + c4);
    }
    wait_async0();
    __syncthreads();
#pragma unroll
    for (int kk = 0; kk < 64; kk += 4) {
      v2f a  = ldA(Xs, mb, kk, lane);
      v2f b0 = ldB(Ws, nb0, kk, lane);
      v2f b1 = ldB(Ws, nb1, kk, lane);
      c0 = wmma4(a, b0, c0);
      c1 = wmma4(a, b1, c1);
    }
    __syncthreads();
  }
#pragma unroll
  for (int r = 0; r < 8; ++r) {
    int m   = m0 + mb + r + 8 * (lane >> 4);
    int nc0 = n0 + nb0 + (lane & 15);
    int nc1 = n0 + nb1 + (lane & 15);
    float v0 = c0[r] + bias[nc0];
    float v1 = c1[r] + bias[nc1];
    if (sig) { v0 = 1.f / (1.f + expf(-v0)); v1 = 1.f / (1.f + expf(-v1)); }
    if (add) { v0 += add[m * DD + nc0]; v1 += add[m * DD + nc1]; }
    out[m * DD + nc0] = v0;
    out[m * DD + nc1] = v1;
  }
}

// ---------------------------------------------------------------------------
// 3-phase chunked linear scan: h_t = a_t*h_{t-1} + b_t ;  hg_t = h_t * g_t
// ---------------------------------------------------------------------------
__global__ void scan_pass1(const float* a, const float* b, float* cA, float* cB) {
  int tid = blockIdx.x * blockDim.x + threadIdx.x;   // B*NCHUNK*D threads
  int d = tid % DD; int rest = tid / DD;
  int c = rest % NCHUNK; int bb = rest / NCHUNK;
  float A = 1.f, Bc = 0.f;
  int base = (bb * TT + c * CL) * DD + d;
  for (int i = 0; i < CL; ++i) {
    float ai = a[base + i * DD], bi = b[base + i * DD];
    A *= ai; Bc = ai * Bc + bi;
  }
  cA[tid] = A; cB[tid] = Bc;
}

__global__ void scan_pass2(const float* cA, const float* cB, float* hinit) {
  int tid = blockIdx.x * blockDim.x + threadIdx.x;   // B*D threads
  int d = tid % DD; int bb = tid / DD;
  float h = 0.f;
  for (int c = 0; c < NCHUNK; ++c) {
    int i = (bb * NCHUNK + c) * DD + d;
    hinit[i] = h;
    h = cA[i] * h + cB[i];
  }
}

__global__ void scan_pass3(const float* a, const float* b, const float* g,
                           const float* hinit, float* hg) {
  int tid = blockIdx.x * blockDim.x + threadIdx.x;
  int d = tid % DD; int rest = tid / DD;
  int c = rest % NCHUNK; int bb = rest / NCHUNK;
  float h = hinit[tid];
  int base = (bb * TT + c * CL) * DD + d;
  for (int i = 0; i < CL; ++i) {
    int idx = base + i * DD;
    h = a[idx] * h + b[idx];
    hg[idx] = h * g[idx];      // may alias 'a' (read a[idx] happens first)
  }
}

// ---------------------------------------------------------------------------
// Chunked decay retention. One block per (batch, head). Per chunk of 64 rows:
//   P = (Q Kc^T) .* decay^(i-j) (causal)      [64x64]
//   O = diag(decay^(i+1)) (Q @ S) + P @ Vc    [64x64]
//   S = decay^64 * S + (diag(decay^(63-j)) Kc)^T @ Vc
// ---------------------------------------------------------------------------
__global__ __launch_bounds__(256) void retention_kernel(
    const float* __restrict__ q, const float* __restrict__ k,
    const float* __restrict__ v, float* __restrict__ ro)
{
  __shared__ float Qs[64 * PITCH], Ks[64 * PITCH], Vs[64 * PITCH];
  __shared__ float Ps[64 * PITCH], Ss[64 * PITCH];
  __shared__ float pw[CL + 1];

  const int tid = threadIdx.x, lane = tid & 31, wave = tid >> 5;
  const int bb = blockIdx.x / HH, hh = blockIdx.x % HH, h0 = hh * DH;
  const int mb = 16 * (wave & 3), nb0 = 32 * (wave >> 2), nb1 = nb0 + 16;
  const uint32_t qs_b = lds_addr32(Qs), ks_b = lds_addr32(Ks), vs_b = lds_addr32(Vs);

  if (tid <= CL) pw[tid] = powf(0.9f, (float)tid);
  for (int i = tid; i < 64 * PITCH; i += 256) Ss[i] = 0.f;
  __syncthreads();

  for (int ch = 0; ch < NCHUNK; ++ch) {
    const int t0 = ch * CL;
    // Stage Q/K/V 64x64 tiles via async global->LDS.
#pragma unroll
    for (int it = 0; it < 4; ++it) {
      int e = tid + 256 * it;
      int r = e >> 4, c4 = (e & 15) << 2;
      size_t g = (size_t)(bb * TT + t0 + r) * DD + h0 + c4;
      uint32_t l = (uint32_t)((r * PITCH + c4) * 4);
      async_b128(qs_b + l, q + g);
      async_b128(ks_b + l, k + g);
      async_b128(vs_b + l, v + g);
    }
    wait_async0();
    __syncthreads();

    // ---- P = Q @ K^T, masked/scaled by decay ----
    v8f p0 = {}, p1 = {};
#pragma unroll
    for (int kk = 0; kk < DH; kk += 4) {
      v2f a = ldA(Qs, mb, kk, lane);
      p0 = wmma4(a, ldBT(Ks, nb0, kk, lane), p0);
      p1 = wmma4(a, ldBT(Ks, nb1, kk, lane), p1);
    }
#pragma unroll
    for (int r = 0; r < 8; ++r) {
      int i  = mb + r + 8 * (lane >> 4);
      int j0 = nb0 + (lane & 15), j1 = nb1 + (lane & 15);
      Ps[i * PITCH + j0] = (i >= j0) ? p0[r] * pw[i - j0] : 0.f;
      Ps[i * PITCH + j1] = (i >= j1) ? p1[r] * pw[i - j1] : 0.f;
    }
    __syncthreads();

    // ---- O = rowscale(Q @ S) + P @ V ----
    v8f o0 = {}, o1 = {};
#pragma unroll
    for (int kk = 0; kk < DH; kk += 4) {
      v2f a = ldA(Qs, mb, kk, lane);
      o0 = wmma4(a, ldB(Ss, nb0, kk, lane), o0);
      o1 = wmma4(a, ldB(Ss, nb1, kk, lane), o1);
    }
#pragma unroll
    for (int r = 0; r < 8; ++r) {
      int i = mb + r + 8 * (lane >> 4);
      float s = pw[i + 1];
      o0[r] *= s; o1[r] *= s;
    }
#pragma unroll
    for (int kk = 0; kk < CL; kk += 4) {
      v2f a = ldA(Ps, mb, kk, lane);
      o0 = wmma4(a, ldB(Vs, nb0, kk, lane), o0);
      o1 = wmma4(a, ldB(Vs, nb1, kk, lane), o1);
    }
#pragma unroll
    for (int r = 0; r < 8; ++r) {
      int i = mb + r + 8 * (lane >> 4);
      int g0 = (bb * TT + t0 + i) * DD + h0 + nb0 + (lane & 15);
      int g1 = (bb * TT + t0 + i) * DD + h0 + nb1 + (lane & 15);
      ro[g0] = o0[r];
      ro[g1] = o1[r];
    }
    __syncthreads();   // all waves done reading S before it is rewritten

    // ---- S = decay^CL * S + (scaled K)^T @ V ----
    v8f s0, s1;
    const float dc = pw[CL];
#pragma unroll
    for (int r = 0; r < 8; ++r) {
      int m = mb + r + 8 * (lane >> 4);
      s0[r] = Ss[m * PITCH + nb0 + (lane & 15)] * dc;
      s1[r] = Ss[m * PITCH + nb1 + (lane & 15)] * dc;
    }
#pragma unroll
    for (int kk = 0; kk < CL; kk += 4) {
      int j = kk + 2 * (lane >> 4);
      int m = mb + (lane & 15);
      v2f a;                               // A[m][j] = K[j][m] * decay^(63-j)
      a.x = Ks[j * PITCH + m]       * pw[63 - j];
      a.y = Ks[(j + 1) * PITCH + m] * pw[62 - j];
      s0 = wmma4(a, ldB(Vs, nb0, kk, lane), s0);
      s1 = wmma4(a, ldB(Vs, nb1, kk, lane), s1);
    }
#pragma unroll
    for (int r = 0; r < 8; ++r) {
      int m = mb + r + 8 * (lane >> 4);
      Ss[m * PITCH + nb0 + (lane & 15)] = s0[r];
      Ss[m * PITCH + nb1 + (lane & 15)] = s1[r];
    }
    __syncthreads();   // before next chunk overwrites Q/K/V and reads new S
  }
}

// ---------------------------------------------------------------------------
extern "C" void kernel_launch(void* const* d_in, const int* in_sizes, int n_in,
                              void* d_out, int out_size, void* d_ws, size_t ws_size,
                              hipStream_t stream) {
  (void)in_sizes; (void)n_in; (void)out_size; (void)ws_size;
  const float* q   = (const float*)d_in[0];
  const float* k   = (const float*)d_in[1];
  const float* v   = (const float*)d_in[2];
  const float* Wa  = (const float*)d_in[3];
  const float* ba  = (const float*)d_in[4];
  const float* Wb  = (const float*)d_in[5];
  const float* bbv = (const float*)d_in[6];
  const float* Wg  = (const float*)d_in[7];
  const float* bg  = (const float*)d_in[8];
  const float* Wo  = (const float*)d_in[9];
  const float* bo  = (const float*)d_in[10];
  float* out = (float*)d_out;

  const size_t S1 = (size_t)BB * TT * DD;          // 4.19M floats
  float* a_buf = (float*)d_ws;                     // reused as hg after scan
  float* b_buf = a_buf + S1;
  float* g_buf = b_buf + S1;
  float* cA    = g_buf + S1;
  float* cB    = cA + (size_t)BB * NCHUNK * DD;
  float* hin   = cB + (size_t)BB * NCHUNK * DD;

  dim3 gg(BB * TT / 64, DD / 64);
  gemm512_kernel<<<gg, 256, 0, stream>>>(q, Wa, ba,  a_buf, nullptr, 1);
  gemm512_kernel<<<gg, 256, 0, stream>>>(q, Wb, bbv, b_buf, nullptr, 0);
  gemm512_kernel<<<gg, 256, 0, stream>>>(q, Wg, bg,  g_buf, nullptr, 1);

  scan_pass1<<<BB * NCHUNK * DD / 256, 256, 0, stream>>>(a_buf, b_buf, cA, cB);
  scan_pass2<<<BB * DD / 256,          256, 0, stream>>>(cA, cB, hin);
  scan_pass3<<<BB * NCHUNK * DD / 256, 256, 0, stream>>>(a_buf, b_buf, g_buf, hin, a_buf);

  retention_kernel<<<BB * HH, 256, 0, stream>>>(q, k, v, out);

  // m_out = hg @ Wo + bo, accumulated on top of retention output already in d_out
  gemm512_kernel<<<gg, 256, 0, stream>>>(a_buf, Wo, bo, out, out, 0);
}